// _ConvBlock4MAG_43018392436810
// MI455X (gfx1250) — compile-verified
//
#include <hip/hip_runtime.h>

// ConvBlock4MAG for MI455X (gfx1250, wave32).
// Strategy: L2-resident scatter via global_atomic_add_f32; dense 32x32 GEMMs
// via V_WMMA_F32_16X16X4_F32 (f32 matrix path keeps reference precision,
// dense part is ~6.5 GFLOP total -> negligible next to the scatter phase).

typedef __attribute__((ext_vector_type(2))) float v2f;
typedef __attribute__((ext_vector_type(8))) float v8f;

#define C32 32

// ---------------------------------------------------------------------------
// Scatter: OUT[b, dst[e], :] += ea[e] * X[b, src[e], :]
// One wave per (edge, batch); lane == channel. Edge metadata is wave-uniform
// (gid>>5 is constant across the wave) so src/dst/ea become scalar loads.
// ---------------------------------------------------------------------------
__global__ void scatter_edges_k(const float* __restrict__ X,
                                const int* __restrict__ src,
                                const int* __restrict__ dst,
                                const float* __restrict__ ea,
                                float* __restrict__ OUT,
                                int nE, int srcN, int dstN) {
  unsigned gid  = blockIdx.x * blockDim.x + threadIdx.x;
  unsigned lane = gid & 31u;
  unsigned wid  = gid >> 5;        // uniform within a wave
  unsigned b    = wid & 3u;        // batch (B == 4)
  unsigned e    = wid >> 2;        // edge index
  if (e >= (unsigned)nE) return;
  int   s = src[e];
  int   d = dst[e];
  float w = ea[e];
  float v = w * X[((size_t)b * (size_t)srcN + (size_t)s) * C32 + lane];
  atomicAdd(&OUT[((size_t)b * (size_t)dstN + (size_t)d) * C32 + lane], v);
}

// ---------------------------------------------------------------------------
// Dense: OUT = X @ Wr + AGG @ Wn + bias   for M x 32 node-feature matrices.
// 256 threads = 8 waves/block; each wave owns a 16-row x 32-col tile.
// Uses V_WMMA_F32_16X16X4_F32 (8 K-steps of 4 over C=32, 2 N-tiles of 16,
// for each of the two input matrices -> 32 WMMAs per wave tile).
//
// Fragment layouts (ISA 7.12.2, wave32):
//   A 16x4 f32 : lanes 0-15 rows M=0..15 with K=0,1 in v0,v1;
//                lanes 16-31 same rows with K=2,3.
//   B 4x16 f32 : lanes 0-15 cols N=0..15 with K=0,1; lanes 16-31 K=2,3.
//   C/D 16x16  : vgpr i -> row i (lanes 0-15) / row i+8 (lanes 16-31).
// ---------------------------------------------------------------------------
__global__ void gconv_dense_k(const float* __restrict__ X,
                              const float* __restrict__ AGG,
                              const float* __restrict__ Wr,
                              const float* __restrict__ Wn,
                              const float* __restrict__ bias,
                              float* __restrict__ OUT, int M) {
  int wave = threadIdx.x >> 5;
  int lane = threadIdx.x & 31;
  int row0 = (blockIdx.x * 8 + wave) * 16;
  if (row0 >= M) return;                 // wave-uniform exit: EXEC stays all-1s

  int r  = lane & 15;                    // row-in-tile for A, col for B/C/D
  int kh = lane >> 4;                    // K-pair select (0 -> K0,1 ; 1 -> K2,3)

  const float* xrow = X   + (size_t)(row0 + r) * C32;
  const float* grow = AGG + (size_t)(row0 + r) * C32;

  v8f acc0 = {};                         // cols 0..15
  v8f acc1 = {};                         // cols 16..31

#pragma unroll
  for (int kk = 0; kk < 8; ++kk) {       // X @ Wr
    int kb = kk * 4 + kh * 2;
    v2f a = *(const v2f*)(xrow + kb);
    v2f b0; b0.x = Wr[kb * C32 + r];       b0.y = Wr[(kb + 1) * C32 + r];
    v2f b1; b1.x = Wr[kb * C32 + 16 + r];  b1.y = Wr[(kb + 1) * C32 + 16 + r];
    acc0 = __builtin_amdgcn_wmma_f32_16x16x4_f32(false, a, false, b0,
                                                 (short)0, acc0, false, false);
    acc1 = __builtin_amdgcn_wmma_f32_16x16x4_f32(false, a, false, b1,
                                                 (short)0, acc1, false, false);
  }
#pragma unroll
  for (int kk = 0; kk < 8; ++kk) {       // AGG @ Wn, accumulated on top
    int kb = kk * 4 + kh * 2;
    v2f a = *(const v2f*)(grow + kb);
    v2f b0; b0.x = Wn[kb * C32 + r];       b0.y = Wn[(kb + 1) * C32 + r];
    v2f b1; b1.x = Wn[kb * C32 + 16 + r];  b1.y = Wn[(kb + 1) * C32 + 16 + r];
    acc0 = __builtin_amdgcn_wmma_f32_16x16x4_f32(false, a, false, b0,
                                                 (short)0, acc0, false, false);
    acc1 = __builtin_amdgcn_wmma_f32_16x16x4_f32(false, a, false, b1,
                                                 (short)0, acc1, false, false);
  }

  float bv0 = bias[r];
  float bv1 = bias[16 + r];
  int   mb  = row0 + kh * 8;             // lanes 16-31 hold rows +8
#pragma unroll
  for (int i = 0; i < 8; ++i) {
    size_t o = (size_t)(mb + i) * C32;
    OUT[o + r]      = acc0[i] + bv0;
    OUT[o + 16 + r] = acc1[i] + bv1;
  }
}

// ---------------------------------------------------------------------------
// Instance norm stats: one block per (b, c) pair (B*C = 128 blocks), reduce
// sum / sumsq over NC nodes, emit mean and rsqrt(var + eps).
// ---------------------------------------------------------------------------
__global__ void inorm_stats_k(const float* __restrict__ X,
                              float* __restrict__ mean,
                              float* __restrict__ rstd, int NCn) {
  int bc = blockIdx.x;                   // 0..127
  int b  = bc >> 5;
  int c  = bc & 31;
  const float* base = X + (size_t)b * (size_t)NCn * C32 + c;
  float s = 0.f, s2 = 0.f;
  for (int n = threadIdx.x; n < NCn; n += blockDim.x) {
    float v = base[(size_t)n * C32];
    s += v;
    s2 += v * v;
  }
  __shared__ float sh[256];
  __shared__ float sh2[256];
  sh[threadIdx.x]  = s;
  sh2[threadIdx.x] = s2;
  __syncthreads();
  for (int off = 128; off > 0; off >>= 1) {
    if ((int)threadIdx.x < off) {
      sh[threadIdx.x]  += sh[threadIdx.x + off];
      sh2[threadIdx.x] += sh2[threadIdx.x + off];
    }
    __syncthreads();
  }
  if (threadIdx.x == 0) {
    float inv = 1.0f / (float)NCn;
    float m   = sh[0] * inv;
    float var = sh2[0] * inv - m * m;
    mean[bc] = m;
    rstd[bc] = rsqrtf(var + 1e-5f);
  }
}

__global__ void inorm_apply_k(float* __restrict__ X,
                              const float* __restrict__ mean,
                              const float* __restrict__ rstd, int NCn) {
  size_t total = (size_t)4 * (size_t)NCn * C32;
  size_t idx = (size_t)blockIdx.x * blockDim.x + threadIdx.x;
  if (idx >= total) return;
  int c  = (int)(idx & 31);
  int b  = (int)(idx / ((size_t)NCn * C32));
  int bc = b * 32 + c;
  X[idx] = (X[idx] - mean[bc]) * rstd[bc];
}

// ---------------------------------------------------------------------------
extern "C" void kernel_launch(void* const* d_in, const int* in_sizes, int n_in,
                              void* d_out, int out_size, void* d_ws, size_t ws_size,
                              hipStream_t stream) {
  const float* x     = (const float*)d_in[0];
  const int*   psrc  = (const int*)d_in[1];
  const int*   pdst  = (const int*)d_in[2];
  const float* pattr = (const float*)d_in[3];
  const int*   esrc  = (const int*)d_in[4];
  const int*   edst  = (const int*)d_in[5];
  const float* eattr = (const float*)d_in[6];
  // d_in[7] = num_coarse (device scalar; derived from out_size instead)
  const float* Wr1 = (const float*)d_in[8];
  const float* Wn1 = (const float*)d_in[9];
  const float* b1  = (const float*)d_in[10];
  const float* Wr2 = (const float*)d_in[11];
  const float* Wn2 = (const float*)d_in[12];
  const float* b2  = (const float*)d_in[13];
  const float* Wr3 = (const float*)d_in[14];
  const float* Wn3 = (const float*)d_in[15];
  const float* b3  = (const float*)d_in[16];
  const float* Wr4 = (const float*)d_in[17];
  const float* Wn4 = (const float*)d_in[18];
  const float* b4  = (const float*)d_in[19];

  const int B  = 4;
  const int EP = in_sizes[1];
  const int E  = in_sizes[4];
  const int NC = out_size / (B * C32);
  const int NF = in_sizes[0] / (B * C32);
  const int M  = B * NC;                 // 200000 rows (multiple of 16)
  const size_t MNC = (size_t)M * C32;    // floats per node-feature matrix

  // Workspace layout (floats): X0 | X1 | AGG | mean(128) | rstd(128)
  float* X0   = (float*)d_ws;
  float* X1   = X0 + MNC;
  float* AGG  = X1 + MNC;
  float* mean = AGG + MNC;
  float* rstd = mean + 128;
  float* out  = (float*)d_out;

  const int TPB = 256;
  int scat_pool_blocks = (int)(((size_t)EP * B * 32 + TPB - 1) / TPB);
  int scat_edge_blocks = (int)(((size_t)E * B * 32 + TPB - 1) / TPB);
  int dense_blocks     = (M / 16 + 7) / 8;
  int norm_blocks      = (int)((MNC + TPB - 1) / TPB);

  // 1. Pooling: X0 = scatter(x, pool edges)
  hipMemsetAsync(X0, 0, MNC * sizeof(float), stream);
  scatter_edges_k<<<scat_pool_blocks, TPB, 0, stream>>>(x, psrc, pdst, pattr,
                                                        X0, EP, NF, NC);
  // 2. Layer 1: X1 = X0 @ Wr1 + agg(X0) @ Wn1 + b1
  hipMemsetAsync(AGG, 0, MNC * sizeof(float), stream);
  scatter_edges_k<<<scat_edge_blocks, TPB, 0, stream>>>(X0, esrc, edst, eattr,
                                                        AGG, E, NC, NC);
  gconv_dense_k<<<dense_blocks, TPB, 0, stream>>>(X0, AGG, Wr1, Wn1, b1, X1, M);

  // 3. Layer 2: X0 = X1 @ Wr2 + agg(X1) @ Wn2 + b2
  hipMemsetAsync(AGG, 0, MNC * sizeof(float), stream);
  scatter_edges_k<<<scat_edge_blocks, TPB, 0, stream>>>(X1, esrc, edst, eattr,
                                                        AGG, E, NC, NC);
  gconv_dense_k<<<dense_blocks, TPB, 0, stream>>>(X1, AGG, Wr2, Wn2, b2, X0, M);

  // 4. Instance norm (in place on X0)
  inorm_stats_k<<<B * C32, TPB, 0, stream>>>(X0, mean, rstd, NC);
  inorm_apply_k<<<norm_blocks, TPB, 0, stream>>>(X0, mean, rstd, NC);

  // 5. Layer 3: X1 = X0 @ Wr3 + agg(X0) @ Wn3 + b3
  hipMemsetAsync(AGG, 0, MNC * sizeof(float), stream);
  scatter_edges_k<<<scat_edge_blocks, TPB, 0, stream>>>(X0, esrc, edst, eattr,
                                                        AGG, E, NC, NC);
  gconv_dense_k<<<dense_blocks, TPB, 0, stream>>>(X0, AGG, Wr3, Wn3, b3, X1, M);

  // 6. Layer 4: out = X1 @ Wr4 + agg(X1) @ Wn4 + b4
  hipMemsetAsync(AGG, 0, MNC * sizeof(float), stream);
  scatter_edges_k<<<scat_edge_blocks, TPB, 0, stream>>>(X1, esrc, edst, eattr,
                                                        AGG, E, NC, NC);
  gconv_dense_k<<<dense_blocks, TPB, 0, stream>>>(X1, AGG, Wr4, Wn4, b4, out, M);
}